// Desends_YsYp_multihead_36507222016278
// MI455X (gfx1250) — compile-verified
//
#include <hip/hip_runtime.h>
#include <hip/hip_bf16.h>
#include <math.h>

typedef __attribute__((ext_vector_type(2))) float v2f;
typedef __attribute__((ext_vector_type(8))) float v8f;

#define DM 256        // D_MODEL
#define HD 32         // HEAD_DIM == wave32 width

// ---------------------------------------------------------------------------
// Kernel 1: segment boundaries from sorted seg_ids.
// seg_start[v] .. seg_start[v+1] is the row range of segment v.
// ---------------------------------------------------------------------------
__global__ void seg_starts_kernel(const int* __restrict__ seg_ids,
                                  int* __restrict__ seg_start, int T, int V) {
  int t = blockIdx.x * blockDim.x + threadIdx.x;
  if (t >= T) return;
  int cur = seg_ids[t];
  if (t == 0) {
    for (int v = 0; v <= cur; ++v) seg_start[v] = 0;
  } else {
    int prev = seg_ids[t - 1];
    for (int v = prev + 1; v <= cur; ++v) seg_start[v] = t;
  }
  if (t == T - 1) {
    for (int v = cur + 1; v <= V; ++v) seg_start[v] = T;
  }
}

// ---------------------------------------------------------------------------
// Kernel 2: segmented online-softmax pooling, single pass over feats.
// One 256-thread block per segment; wave h owns head h (32 lanes == HEAD_DIM).
// Both ys and yp pooled in the same pass -> feats read exactly once (~819 MB).
// ---------------------------------------------------------------------------
__global__ __launch_bounds__(256) void pool_kernel(
    const float* __restrict__ feats, const int* __restrict__ seg_start,
    const float* __restrict__ ys_attn, const float* __restrict__ yp_attn,
    float* __restrict__ pooled_ys, float* __restrict__ pooled_yp) {
  const int v    = blockIdx.x;
  const int col  = threadIdx.x;          // = head*32 + lane
  const float a_s = ys_attn[col];
  const float a_p = yp_attn[col];
  const int start = seg_start[v];
  const int end   = seg_start[v + 1];

  float m_s = -INFINITY, m_p = -INFINITY;
  float d_s = 0.f, d_p = 0.f;
  float acc_s = 0.f, acc_p = 0.f;

  for (int t = start; t < end; ++t) {
    const float x = feats[(size_t)t * DM + col];
    // per-head logits: reduce x*attn over the 32 lanes of this wave
    float ls = x * a_s;
    float lp = x * a_p;
#pragma unroll
    for (int o = 16; o > 0; o >>= 1) {
      ls += __shfl_xor(ls, o, 32);
      lp += __shfl_xor(lp, o, 32);
    }
    // online softmax update (ys)
    float mn = fmaxf(m_s, ls);
    float sc = __expf(m_s - mn);   // first iter: exp(-inf)=0
    float e  = __expf(ls - mn);
    d_s   = d_s * sc + e;
    acc_s = acc_s * sc + e * x;
    m_s   = mn;
    // online softmax update (yp)
    mn = fmaxf(m_p, lp);
    sc = __expf(m_p - mn);
    e  = __expf(lp - mn);
    d_p   = d_p * sc + e;
    acc_p = acc_p * sc + e * x;
    m_p   = mn;
  }

  const size_t o = (size_t)v * DM + col;
  if (end > start) {
    pooled_ys[o] = acc_s / d_s;
    pooled_yp[o] = acc_p / d_p;
  } else {
    pooled_ys[o] = 0.f;
    pooled_yp[o] = 0.f;
  }
}

// ---------------------------------------------------------------------------
// Kernel 3: out[v, n] = sum_k A[v,k] * W[n,k] + bias[n]  (out = A @ W^T + b)
// Full-precision fp32 WMMA: V_WMMA_F32_16X16X4_F32.
// Each wave computes a 16(M) x 64(N) block: A fragment reused across 4
// accumulators; K=256 consumed in 64 steps of 4.
// Fragment layout per CDNA5 ISA (32-bit A/B, 16x4 / 4x16):
//   lanes 0-15:  v0=K0, v1=K1 ; lanes 16-31: v0=K2, v1=K3  -> float2 loads.
// ---------------------------------------------------------------------------
__global__ __launch_bounds__(256) void gemm_bias_kernel(
    const float* __restrict__ A,     // [V][256] pooled
    const float* __restrict__ W,     // [256][256] row-major, W[n][k]
    const float* __restrict__ bias,  // [256]
    float* __restrict__ out,         // [V][256]
    int V) {
  const int wavesPerBlock = blockDim.x >> 5;
  const int waveId  = blockIdx.x * wavesPerBlock + (threadIdx.x >> 5);
  const int nRowTiles = V >> 4;
  const int rowTile = waveId >> 2;      // 4 column-quads (64 cols each) per row tile
  const int colQuad = waveId & 3;
  if (rowTile >= nRowTiles) return;

  const int lane  = threadIdx.x & 31;
  const int lhalf = lane >> 4;          // 0: K=k0..k0+1, 1: K=k0+2..k0+3
  const int l16   = lane & 15;
  const int koff  = lhalf * 2;
  const int m     = rowTile * 16 + l16; // A row for this lane
  const int n0    = colQuad * 64;
  const int nB    = n0 + l16;           // base B column for this lane

  const float* Arow  = A + (size_t)m * DM + koff;
  const float* Wbase = W + koff;

  v8f c[4];
  c[0] = (v8f){}; c[1] = (v8f){}; c[2] = (v8f){}; c[3] = (v8f){};

  for (int k0 = 0; k0 < DM; k0 += 4) {
    v2f a  = *(const v2f*)(Arow + k0);
    v2f b0 = *(const v2f*)(Wbase + (size_t)(nB +  0) * DM + k0);
    v2f b1 = *(const v2f*)(Wbase + (size_t)(nB + 16) * DM + k0);
    v2f b2 = *(const v2f*)(Wbase + (size_t)(nB + 32) * DM + k0);
    v2f b3 = *(const v2f*)(Wbase + (size_t)(nB + 48) * DM + k0);
    // 8 args: (neg_a, A, neg_b, B, c_mod, C, reuse_a, reuse_b)
    c[0] = __builtin_amdgcn_wmma_f32_16x16x4_f32(false, a, false, b0, (short)0, c[0], false, false);
    c[1] = __builtin_amdgcn_wmma_f32_16x16x4_f32(false, a, false, b1, (short)0, c[1], false, false);
    c[2] = __builtin_amdgcn_wmma_f32_16x16x4_f32(false, a, false, b2, (short)0, c[2], false, false);
    c[3] = __builtin_amdgcn_wmma_f32_16x16x4_f32(false, a, false, b3, (short)0, c[3], false, false);
  }

  // C/D layout: VGPR r -> row = 16*rowTile + 8*lhalf + r, col = tile_n0 + l16
  const int rowBase = rowTile * 16 + lhalf * 8;
#pragma unroll
  for (int q = 0; q < 4; ++q) {
    const int n  = n0 + q * 16 + l16;
    const float bv = bias[n];
    v8f cq = c[q];
#pragma unroll
    for (int r = 0; r < 8; ++r) {
      out[(size_t)(rowBase + r) * DM + n] = cq[r] + bv;
    }
  }
}

// ---------------------------------------------------------------------------
// Launch
// ---------------------------------------------------------------------------
extern "C" void kernel_launch(void* const* d_in, const int* in_sizes, int n_in,
                              void* d_out, int out_size, void* d_ws, size_t ws_size,
                              hipStream_t stream) {
  const float* feats   = (const float*)d_in[0];
  const int*   seg_ids = (const int*)d_in[1];
  const float* ys_attn = (const float*)d_in[2];
  const float* yp_attn = (const float*)d_in[3];
  const float* W_ys    = (const float*)d_in[4];
  const float* b_ys    = (const float*)d_in[5];
  const float* W_yp    = (const float*)d_in[6];
  const float* b_yp    = (const float*)d_in[7];
  float* out = (float*)d_out;

  const int T = in_sizes[1];           // rows
  const int V = out_size / (2 * DM);   // segments

  // workspace layout: seg_start (V+1 ints) | pooled_ys | pooled_yp
  char* ws = (char*)d_ws;
  int*   seg_start = (int*)ws;
  size_t ofs = ((size_t)(V + 1) * sizeof(int) + 255) & ~(size_t)255;
  float* pooled_ys = (float*)(ws + ofs);
  float* pooled_yp = pooled_ys + (size_t)V * DM;

  seg_starts_kernel<<<(T + 255) / 256, 256, 0, stream>>>(seg_ids, seg_start, T, V);

  pool_kernel<<<V, 256, 0, stream>>>(feats, seg_start, ys_attn, yp_attn,
                                     pooled_ys, pooled_yp);

  const int wavesNeeded = (V / 16) * 4;
  const int blocks = (wavesNeeded + 7) / 8;   // 8 waves / 256-thread block
  gemm_bias_kernel<<<blocks, 256, 0, stream>>>(pooled_ys, W_ys, b_ys,
                                               out, V);
  gemm_bias_kernel<<<blocks, 256, 0, stream>>>(pooled_yp, W_yp, b_yp,
                                               out + (size_t)V * DM, V);
}